// QonvLayer_65472481460760
// MI455X (gfx1250) — compile-verified
//
#include <hip/hip_runtime.h>
#include <cstdint>
#include <cmath>

typedef float v2f __attribute__((ext_vector_type(2)));
typedef float v8f __attribute__((ext_vector_type(8)));

#define NSTEPS 21   // 84 = 21 * 4 K-steps (81 monomials padded)

struct GateList { int n; int code[96]; };

// ---------------- host-side numpy RandomState(42) replication ----------------
struct MT19937 {
  uint32_t mt[624]; int idx;
  void seed(uint32_t s) {
    mt[0] = s;
    for (int i = 1; i < 624; ++i)
      mt[i] = 1812433253u * (mt[i - 1] ^ (mt[i - 1] >> 30)) + (uint32_t)i;
    idx = 624;
  }
  uint32_t next32() {
    if (idx >= 624) {
      for (int i = 0; i < 624; ++i) {
        uint32_t y = (mt[i] & 0x80000000u) | (mt[(i + 1) % 624] & 0x7fffffffu);
        uint32_t v = y >> 1;
        if (y & 1u) v ^= 2567483615u;
        mt[i] = mt[(i + 397) % 624] ^ v;
      }
      idx = 0;
    }
    uint32_t y = mt[idx++];
    y ^= y >> 11;
    y ^= (y << 7)  & 2636928640u;
    y ^= (y << 15) & 4022730752u;
    y ^= y >> 18;
    return y;
  }
  double random_sample() {  // numpy rk_double: two u32 draws
    uint32_t a = next32() >> 5, b = next32() >> 6;
    return (a * 67108864.0 + b) / 9007199254740992.0;
  }
  // numpy legacy masked rejection over [0, rmax] inclusive (32-bit path)
  uint32_t bounded(uint32_t rmax) {
    if (rmax == 0) return 0;
    uint32_t mask = rmax;
    mask |= mask >> 1; mask |= mask >> 2; mask |= mask >> 4;
    mask |= mask >> 8; mask |= mask >> 16;
    uint32_t v;
    do { v = next32() & mask; } while (v > rmax);
    return v;
  }
};

static void build_gates(GateList& gl) {
  MT19937 rng; rng.seed(42);
  gl.n = 0;
  for (int l = 0; l < 4; ++l) {
    int i = 0, guard = 0;
    while (i < 8 && guard < 4096) {
      ++guard;
      if (rng.random_sample() > 0.3) {
        int g = (int)rng.bounded(2);  // randint(3)
        int w = (int)rng.bounded(3);  // randint(4)
        if (gl.n < 96) gl.code[gl.n++] = 0 | (g << 1) | (w << 5) | ((l * 8 + i) << 9);
        ++i;
      } else {                        // choice(4, 2, replace=False) == permutation(4)[:2]
        int arr[4] = {0, 1, 2, 3};
        for (int k = 3; k >= 1; --k) {
          int j = (int)rng.bounded((uint32_t)k);
          int t = arr[k]; arr[k] = arr[j]; arr[j] = t;
        }
        if (gl.n < 96) gl.code[gl.n++] = 1 | (arr[0] << 1) | (arr[1] << 5);
      }
    }
  }
}

// ---------------- device kernels ----------------

__global__ __launch_bounds__(256) void zero_kernel(float4* __restrict__ p, int n4) {
  int i = blockIdx.x * blockDim.x + threadIdx.x;
  if (i < n4) p[i] = make_float4(0.f, 0.f, 0.f, 0.f);
}

// One wave. Lanes 0..15 each simulate the fixed circuit on basis column `lane`,
// producing W (16x16 complex) in LDS; then all 32 lanes fold W into the
// 84x16 monomial-coefficient matrix Cmat (rows 81..83 and cols 4..15 zero).
__global__ __launch_bounds__(32) void setup_kernel(const float* __restrict__ weights,
                                                   float* __restrict__ Cmat,
                                                   GateList gl) {
  __shared__ float wr[256], wi[256];
  __shared__ float Cl[84 * 16];
  const int lane = threadIdx.x;

  for (int i = lane; i < 84 * 16; i += 32) Cl[i] = 0.f;

  if (lane < 16) {
    for (int k = 0; k < 16; ++k) {
      wr[lane * 16 + k] = (k == lane) ? 1.f : 0.f;
      wi[lane * 16 + k] = 0.f;
    }
    for (int gi = 0; gi < gl.n; ++gi) {
      const int code = gl.code[gi];
      const int kind = code & 1;
      const int p0 = (code >> 1) & 15;
      const int p1 = (code >> 5) & 15;
      if (kind == 0) {  // rotation gate: g=p0, wire=p1
        const int aidx = (code >> 9) & 255;
        const float h = 0.5f * weights[aidx];
        const float c = cosf(h), s = sinf(h);
        const int bit = 1 << (3 - p1);
        for (int k = 0; k < 16; ++k) {
          if (k & bit) continue;
          const int k1 = k | bit;
          const float a0r = wr[lane * 16 + k],  a0i = wi[lane * 16 + k];
          const float a1r = wr[lane * 16 + k1], a1i = wi[lane * 16 + k1];
          float n0r, n0i, n1r, n1i;
          if (p0 == 0) {        // RX: [[c,-is],[-is,c]]
            n0r = c * a0r + s * a1i;  n0i = c * a0i - s * a1r;
            n1r = c * a1r + s * a0i;  n1i = c * a1i - s * a0r;
          } else if (p0 == 1) { // RY: [[c,-s],[s,c]]
            n0r = c * a0r - s * a1r;  n0i = c * a0i - s * a1i;
            n1r = s * a0r + c * a1r;  n1i = s * a0i + c * a1i;
          } else {              // RZ: diag(c-is, c+is)
            n0r = c * a0r + s * a0i;  n0i = c * a0i - s * a0r;
            n1r = c * a1r - s * a1i;  n1i = c * a1i + s * a1r;
          }
          wr[lane * 16 + k]  = n0r; wi[lane * 16 + k]  = n0i;
          wr[lane * 16 + k1] = n1r; wi[lane * 16 + k1] = n1i;
        }
      } else {  // CNOT: control p0, target p1
        const int cb = 1 << (3 - p0), tb = 1 << (3 - p1);
        for (int k = 0; k < 16; ++k) {
          if ((k & cb) && !(k & tb)) {
            const int k1 = k | tb;
            float t;
            t = wr[lane * 16 + k]; wr[lane * 16 + k] = wr[lane * 16 + k1]; wr[lane * 16 + k1] = t;
            t = wi[lane * 16 + k]; wi[lane * 16 + k] = wi[lane * 16 + k1]; wi[lane * 16 + k1] = t;
          }
        }
      }
    }
  }
  __syncthreads();

  // C[m, ch] = sum over (k,l) with matching ternary digits of
  //            Re( sum_idx signs[ch,idx] * conj(W[idx,k]) * W[idx,l] )
  // W[idx, col] is stored at wr/wi[col*16 + idx].
  for (int p = lane; p < 256; p += 32) {
    const int k = p >> 4, l = p & 15;
    float sum[4] = {0.f, 0.f, 0.f, 0.f};
    for (int idx = 0; idx < 16; ++idx) {
      const float v = wr[k * 16 + idx] * wr[l * 16 + idx] +
                      wi[k * 16 + idx] * wi[l * 16 + idx];
      #pragma unroll
      for (int ch = 0; ch < 4; ++ch) {
        const float sg = ((idx >> (3 - ch)) & 1) ? -1.f : 1.f;
        sum[ch] += sg * v;
      }
    }
    int m = 0;
    #pragma unroll
    for (int j = 0; j < 4; ++j) {
      const int d = ((k >> (3 - j)) & 1) + ((l >> (3 - j)) & 1);
      m = m * 3 + d;
    }
    #pragma unroll
    for (int ch = 0; ch < 4; ++ch) atomicAdd(&Cl[m * 16 + ch], sum[ch]);
  }
  __syncthreads();

  for (int i = lane; i < 84 * 16; i += 32) Cmat[i] = Cl[i];
}

// One wave per tile of 16 patches (16 consecutive kk). Lanes L and L+16 share a
// patch: A-fragment of V_WMMA_F32_16X16X4_F32 puts K=0,1 in lanes 0-15 and
// K=2,3 in lanes 16-31. D = (16 patches) x (16 cols; cols 0..3 = channels).
__global__ __launch_bounds__(256) void qonv_main(const float* __restrict__ img,
                                                 const float* __restrict__ Cmat,
                                                 float* __restrict__ out) {
  const int lane = threadIdx.x & 31;
  const int wave = threadIdx.x >> 5;
  const int tile = blockIdx.x * 8 + wave;      // 0..32767
  const int b    = tile >> 10;                 // 128*8 = 1024 tiles per image
  const int rem  = tile & 1023;
  const int jj   = rem >> 3;                   // 0..127
  const int tkk  = rem & 7;                    // 0..7

  const int col16 = lane & 15;
  const int half  = lane >> 4;

  const int kk = tkk * 16 + col16;
  const int r0 = 2 * jj;
  const int c0 = 2 * kk;

  const float* p0 = img + (((b * 512) + r0) * 512 + c0) * 3;
  const float* p1 = img + (((b * 512) + r0 + 1) * 512 + c0) * 3;
  const float inv3 = 1.0f / 3.0f;

  float th[4];
  th[0] = (p0[0] + p0[1] + p0[2]) * inv3;   // wire 0: (r, c)
  th[1] = (p0[3] + p0[4] + p0[5]) * inv3;   // wire 1: (r, c+1)
  th[2] = (p1[0] + p1[1] + p1[2]) * inv3;   // wire 2: (r+1, c)
  th[3] = (p1[3] + p1[4] + p1[5]) * inv3;   // wire 3: (r+1, c+1)

  float q[4][3];
  #pragma unroll
  for (int w = 0; w < 4; ++w) {
    const float h = 0.5f * th[w];
    const float c = __cosf(h), s = __sinf(h);
    q[w][0] = c * c; q[w][1] = c * s; q[w][2] = s * s;
  }

  // triple products over wires 0..2; pre[27] = 0 pads monomials 81..83
  float pre[28];
  #pragma unroll
  for (int a = 0; a < 3; ++a)
    #pragma unroll
    for (int bb = 0; bb < 3; ++bb)
      #pragma unroll
      for (int cc = 0; cc < 3; ++cc)
        pre[(a * 3 + bb) * 3 + cc] = q[0][a] * q[1][bb] * q[2][cc];
  pre[27] = 0.0f;

  v8f acc = {0.f, 0.f, 0.f, 0.f, 0.f, 0.f, 0.f, 0.f};
  const int kboff = 2 * half;

  #pragma unroll
  for (int t = 0; t < NSTEPS; ++t) {
    const int m0 = 4 * t;
    // all pre[]/q[] indices are compile-time constants after unrolling
    const float lo0 = pre[(m0 + 0) / 3] * q[3][(m0 + 0) % 3];
    const float lo1 = pre[(m0 + 1) / 3] * q[3][(m0 + 1) % 3];
    const float hi0 = pre[(m0 + 2) / 3] * q[3][(m0 + 2) % 3];
    const float hi1 = pre[(m0 + 3) / 3] * q[3][(m0 + 3) % 3];
    v2f A;
    A.x = half ? hi0 : lo0;
    A.y = half ? hi1 : lo1;
    v2f B;
    B.x = Cmat[(m0 + kboff + 0) * 16 + col16];
    B.y = Cmat[(m0 + kboff + 1) * 16 + col16];
    acc = __builtin_amdgcn_wmma_f32_16x16x4_f32(false, A, false, B, (short)0, acc,
                                                false, false);
  }

  // D layout: lanes 0-15 hold N=lane, M=vgpr r (0..7); lanes 16-31: N=lane-16, M=8+r.
  const int N = col16;
  if (N < 4) {
    const int mbase = half * 8;
    const int rowbase = ((b * 256) + jj) * 256;
    #pragma unroll
    for (int r = 0; r < 8; ++r) {
      const int kko = tkk * 16 + mbase + r;
      out[(rowbase + kko) * 4 + N] = acc[r];
    }
  }
}

extern "C" void kernel_launch(void* const* d_in, const int* in_sizes, int n_in,
                              void* d_out, int out_size, void* d_ws, size_t ws_size,
                              hipStream_t stream) {
  (void)in_sizes; (void)n_in; (void)ws_size;
  const float* img     = (const float*)d_in[0];   // (32,512,512,3) f32
  const float* weights = (const float*)d_in[1];   // (4,8) f32
  float* out  = (float*)d_out;                    // (32,256,256,4) f32
  float* Cmat = (float*)d_ws;                     // 84*16 f32 scratch

  GateList gl;
  build_gates(gl);

  const int n4 = out_size / 4;  // out_size = 8,388,608 floats
  zero_kernel<<<(n4 + 255) / 256, 256, 0, stream>>>((float4*)d_out, n4);
  setup_kernel<<<1, 32, 0, stream>>>(weights, Cmat, gl);
  qonv_main<<<4096, 256, 0, stream>>>(img, Cmat, out);
}